// SATGraphNN_73272142070153
// MI455X (gfx1250) — compile-verified
//
#include <hip/hip_runtime.h>
#include <hip/hip_bf16.h>

typedef float v2f __attribute__((ext_vector_type(2)));
typedef float v8f __attribute__((ext_vector_type(8)));

// ---------------------------------------------------------------------------
// Fused per-edge MLP + attention scale + scatter-add.
// One wave (32 lanes, wave32) handles a tile of 16 edges.
//   layer1: h[16,64] = relu(x[16,5] @ w1.T + b1)   (VALU, computed directly in
//           WMMA A-fragment layout: lane l<16 holds row l K={4k,4k+1},
//           lane l+16 holds K={4k+2,4k+3})
//   layer2: m[16,64] = h @ w2.T + b2               (v_wmma_f32_16x16x4_f32)
//   m *= att[scatter_idx[e]]  (optional)
//   atomicAdd into outMsg[scatter_idx[e]*64 + n]
// ---------------------------------------------------------------------------
__global__ __launch_bounds__(256)
void edge_mlp_scatter_kernel(const float* __restrict__ nodeFeat,   // [*,3] gathered side
                             const int*   __restrict__ gidx,       // [E] gather index
                             const float* __restrict__ ef,         // [E,2]
                             const float* __restrict__ w1,         // [64,5]
                             const float* __restrict__ b1,         // [64]
                             const float* __restrict__ w2,         // [64,64]
                             const float* __restrict__ b2,         // [64]
                             const float* __restrict__ attExp,     // [C] or null
                             const float* __restrict__ invS,       // [1]
                             const int*   __restrict__ sidx,       // [E] scatter index
                             float*       __restrict__ outMsg,     // [*,64]
                             int E, int numTiles)
{
    __shared__ float w1s[64 * 5];
    __shared__ float b1s[64];
    __shared__ float w2s[64 * 64];
    __shared__ float b2s[64];

    const int tid = threadIdx.x;
    for (int i = tid; i < 64 * 5; i += 256) w1s[i] = w1[i];
    for (int i = tid; i < 64; i += 256) { b1s[i] = b1[i]; b2s[i] = b2[i]; }
    for (int i = tid; i < 64 * 64; i += 256) w2s[i] = w2[i];
    __syncthreads();

    const int lane = tid & 31;
    // Force tile into an SGPR: all tile-derived branches become scalar.
    const int tile = __builtin_amdgcn_readfirstlane(blockIdx.x * 8 + (tid >> 5));
    if (tile >= numTiles) return;               // scalar branch, whole wave

    const int r  = lane & 15;                   // row (edge within tile) / col N
    const int hi = lane >> 4;                   // half selector for K
    const int e  = tile * 16 + r;
    const bool fullTile = (tile * 16 + 16 <= E);   // scalar

    float x0 = 0.f, x1 = 0.f, x2 = 0.f, x3 = 0.f, x4 = 0.f;
    int   sI = 0;
    float aw = 1.0f;
    if (fullTile || e < E) {
        const int g = gidx[e];
        x0 = nodeFeat[g * 3 + 0];
        x1 = nodeFeat[g * 3 + 1];
        x2 = nodeFeat[g * 3 + 2];
        x3 = ef[(size_t)e * 2 + 0];
        x4 = ef[(size_t)e * 2 + 1];
        sI = sidx[e];
        if (attExp) aw = attExp[sI] * invS[0];
    }

    // ---- layer 1 in A-fragment layout: lane holds h[row r][4k+2*hi + {0,1}]
    v2f hfrag[16];
#pragma unroll
    for (int kk = 0; kk < 16; ++kk) {
        const int j0 = kk * 4 + hi * 2;
        const int j1 = j0 + 1;
        float s0 = b1s[j0] + x0 * w1s[j0 * 5 + 0] + x1 * w1s[j0 * 5 + 1]
                 + x2 * w1s[j0 * 5 + 2] + x3 * w1s[j0 * 5 + 3] + x4 * w1s[j0 * 5 + 4];
        float s1 = b1s[j1] + x0 * w1s[j1 * 5 + 0] + x1 * w1s[j1 * 5 + 1]
                 + x2 * w1s[j1 * 5 + 2] + x3 * w1s[j1 * 5 + 3] + x4 * w1s[j1 * 5 + 4];
        hfrag[kk].x = fmaxf(s0, 0.f);
        hfrag[kk].y = fmaxf(s1, 0.f);
    }

    // ---- broadcast per-row scatter index / attention weight (nt-invariant)
    int   sIm[8];
    float awm[8];
#pragma unroll
    for (int d = 0; d < 8; ++d) {
        const int m = hi * 8 + d;               // D: VGPR d -> row m = d + 8*hi
        sIm[d] = __shfl(sI, m, 32);
        awm[d] = __shfl(aw, m, 32);
    }

    // ---- layer 2: 4 N-tiles of 16 columns, K=64 via 16 x (16x16x4 fp32 WMMA)
    for (int nt = 0; nt < 4; ++nt) {
        const int n = nt * 16 + r;              // this lane's output column
        v8f acc = {0.f, 0.f, 0.f, 0.f, 0.f, 0.f, 0.f, 0.f};
#pragma unroll
        for (int kk = 0; kk < 16; ++kk) {
            v2f bfrag;
            const int off = n * 64 + kk * 4 + hi * 2;   // B: lane=N, regs span K
            bfrag.x = w2s[off];
            bfrag.y = w2s[off + 1];
            acc = __builtin_amdgcn_wmma_f32_16x16x4_f32(
                false, hfrag[kk], false, bfrag, (short)0, acc, false, false);
        }
        const float bn = b2s[n];
        if (fullTile) {                         // scalar branch: common case
#pragma unroll
            for (int d = 0; d < 8; ++d) {
                atomicAdd(&outMsg[(size_t)sIm[d] * 64 + n], (acc[d] + bn) * awm[d]);
            }
        } else {                                // tail tile (rare)
#pragma unroll
            for (int d = 0; d < 8; ++d) {
                if (tile * 16 + hi * 8 + d < E)
                    atomicAdd(&outMsg[(size_t)sIm[d] * 64 + n], (acc[d] + bn) * awm[d]);
            }
        }
    }
}

// ---------------------------------------------------------------------------
// Attention MLP over clauses: attExp[c] = exp(tanh(cf@w1.T+b1)@w2.T+b2),
// plus deterministic per-block partial sums for the softmax denominator.
// ---------------------------------------------------------------------------
__global__ __launch_bounds__(256)
void att_kernel(const float* __restrict__ cf,      // [C,3]
                const float* __restrict__ w1,      // [64,3]
                const float* __restrict__ b1,      // [64]
                const float* __restrict__ w2,      // [64]
                const float* __restrict__ b2,      // [1]
                float* __restrict__ attExp,        // [C]
                float* __restrict__ partials, int C)
{
    __shared__ float w1s[192], b1s[64], w2s[64];
    __shared__ float red[256];
    const int tid = threadIdx.x;
    for (int i = tid; i < 192; i += 256) w1s[i] = w1[i];
    if (tid < 64) { b1s[tid] = b1[tid]; w2s[tid] = w2[tid]; }
    __syncthreads();

    const int c = blockIdx.x * 256 + tid;
    float eo = 0.f;
    if (c < C) {
        const float c0 = cf[(size_t)c * 3 + 0];
        const float c1 = cf[(size_t)c * 3 + 1];
        const float c2 = cf[(size_t)c * 3 + 2];
        float a = b2[0];
#pragma unroll 4
        for (int j = 0; j < 64; ++j) {
            const float t = tanhf(b1s[j] + c0 * w1s[j * 3] + c1 * w1s[j * 3 + 1]
                                         + c2 * w1s[j * 3 + 2]);
            a += t * w2s[j];
        }
        eo = __expf(a);
        attExp[c] = eo;
    }
    red[tid] = eo;
    __syncthreads();
    for (int s = 128; s > 0; s >>= 1) {
        if (tid < s) red[tid] += red[tid + s];
        __syncthreads();
    }
    if (tid == 0) partials[blockIdx.x] = red[0];
}

__global__ __launch_bounds__(256)
void reduce_inv_kernel(const float* __restrict__ partials, int n,
                       float* __restrict__ invS)
{
    __shared__ float red[256];
    float s = 0.f;
    for (int i = threadIdx.x; i < n; i += 256) s += partials[i];
    red[threadIdx.x] = s;
    __syncthreads();
    for (int st = 128; st > 0; st >>= 1) {
        if (threadIdx.x < st) red[threadIdx.x] += red[threadIdx.x + st];
        __syncthreads();
    }
    if (threadIdx.x == 0) invS[0] = 1.0f / red[0];
}

// ---------------------------------------------------------------------------
// GRUCell with input=64, hidden=3 (PyTorch semantics). One thread per row.
// msg rows are 256B-aligned -> float4 loads.
// ---------------------------------------------------------------------------
__global__ __launch_bounds__(256)
void gru_kernel(const float* __restrict__ msg,   // [N,64]
                const float* __restrict__ h,     // [N,3]
                const float* __restrict__ wih,   // [9,64]
                const float* __restrict__ whh,   // [9,3]
                const float* __restrict__ bih,   // [9]
                const float* __restrict__ bhh,   // [9]
                float* __restrict__ hout,        // [N,3]
                int N)
{
    __shared__ float wihs[576];
    __shared__ float whhs[27], bihs[9], bhhs[9];
    const int tid = threadIdx.x;
    for (int i = tid; i < 576; i += 256) wihs[i] = wih[i];
    if (tid < 27) whhs[tid] = whh[tid];
    if (tid < 9) { bihs[tid] = bih[tid]; bhhs[tid] = bhh[tid]; }
    __syncthreads();

    const int n = blockIdx.x * 256 + tid;
    if (n >= N) return;

    float gi[9];
#pragma unroll
    for (int g = 0; g < 9; ++g) gi[g] = bihs[g];
    const float4* x4 = reinterpret_cast<const float4*>(msg + (size_t)n * 64);
#pragma unroll 4
    for (int k4 = 0; k4 < 16; ++k4) {
        const float4 xv = x4[k4];
#pragma unroll
        for (int g = 0; g < 9; ++g) {
            gi[g] += xv.x * wihs[g * 64 + k4 * 4 + 0];
            gi[g] += xv.y * wihs[g * 64 + k4 * 4 + 1];
            gi[g] += xv.z * wihs[g * 64 + k4 * 4 + 2];
            gi[g] += xv.w * wihs[g * 64 + k4 * 4 + 3];
        }
    }
    const float h0 = h[(size_t)n * 3 + 0];
    const float h1 = h[(size_t)n * 3 + 1];
    const float h2 = h[(size_t)n * 3 + 2];
    const float hh[3] = {h0, h1, h2};
#pragma unroll
    for (int i = 0; i < 3; ++i) {
        const float ghr = bhhs[i]     + h0 * whhs[i * 3]       + h1 * whhs[i * 3 + 1]       + h2 * whhs[i * 3 + 2];
        const float ghz = bhhs[3 + i] + h0 * whhs[(3 + i) * 3] + h1 * whhs[(3 + i) * 3 + 1] + h2 * whhs[(3 + i) * 3 + 2];
        const float ghn = bhhs[6 + i] + h0 * whhs[(6 + i) * 3] + h1 * whhs[(6 + i) * 3 + 1] + h2 * whhs[(6 + i) * 3 + 2];
        const float rr = 1.f / (1.f + __expf(-(gi[i] + ghr)));
        const float zz = 1.f / (1.f + __expf(-(gi[3 + i] + ghz)));
        const float nn = tanhf(gi[6 + i] + rr * ghn);
        hout[(size_t)n * 3 + i] = (1.f - zz) * nn + zz * hh[i];
    }
}

// n must be a multiple of 4 (we only zero [*,64] buffers).
__global__ __launch_bounds__(256)
void zero_kernel(float4* __restrict__ p, size_t n4)
{
    size_t i = (size_t)blockIdx.x * blockDim.x + threadIdx.x;
    const size_t stride = (size_t)gridDim.x * blockDim.x;
    const float4 z = {0.f, 0.f, 0.f, 0.f};
    for (; i < n4; i += stride) p[i] = z;
}

// ---------------------------------------------------------------------------
extern "C" void kernel_launch(void* const* d_in, const int* in_sizes, int n_in,
                              void* d_out, int out_size, void* d_ws, size_t ws_size,
                              hipStream_t stream)
{
    (void)n_in; (void)out_size; (void)ws_size;
    const float* var_feat = (const float*)d_in[0];
    const float* cls_feat = (const float*)d_in[1];
    const float* ef       = (const float*)d_in[2];
    const float* v2c_w1   = (const float*)d_in[3];
    const float* v2c_b1   = (const float*)d_in[4];
    const float* v2c_w2   = (const float*)d_in[5];
    const float* v2c_b2   = (const float*)d_in[6];
    const float* c2v_w1   = (const float*)d_in[7];
    const float* c2v_b1   = (const float*)d_in[8];
    const float* c2v_w2   = (const float*)d_in[9];
    const float* c2v_b2   = (const float*)d_in[10];
    const float* var_wih  = (const float*)d_in[11];
    const float* var_whh  = (const float*)d_in[12];
    const float* var_bih  = (const float*)d_in[13];
    const float* var_bhh  = (const float*)d_in[14];
    const float* cls_wih  = (const float*)d_in[15];
    const float* cls_whh  = (const float*)d_in[16];
    const float* cls_bih  = (const float*)d_in[17];
    const float* cls_bhh  = (const float*)d_in[18];
    const float* att_w1   = (const float*)d_in[19];
    const float* att_b1   = (const float*)d_in[20];
    const float* att_w2   = (const float*)d_in[21];
    const float* att_b2   = (const float*)d_in[22];
    const int*   var_idx  = (const int*)d_in[23];
    const int*   cls_idx  = (const int*)d_in[24];

    const int V = in_sizes[0] / 3;
    const int C = in_sizes[1] / 3;
    const int E = in_sizes[23];
    const int L = 3;

    // workspace layout (cmsg/vmsg share one buffer; consumed before reuse)
    char* ws = (char*)d_ws;
    size_t off = 0;
    auto walloc = [&](size_t bytes) -> void* {
        void* p = ws + off;
        off += (bytes + 255) & ~(size_t)255;
        return p;
    };
    float* msg      = (float*)walloc((size_t)C * 64 * sizeof(float));
    float* attExp   = (float*)walloc((size_t)C * sizeof(float));
    float* partials = (float*)walloc(4096 * sizeof(float));
    float* invS     = (float*)walloc(256);
    float* vfA      = (float*)walloc((size_t)V * 3 * sizeof(float));
    float* vfB      = (float*)walloc((size_t)V * 3 * sizeof(float));
    float* cfA      = (float*)walloc((size_t)C * 3 * sizeof(float));
    float* cfB      = (float*)walloc((size_t)C * 3 * sizeof(float));

    float* outV = (float*)d_out;                   // rows [0,V)
    float* outC = (float*)d_out + (size_t)V * 3;   // rows [V,V+C)

    const float* vf_cur = var_feat;
    const float* cf_cur = cls_feat;

    const int numTiles = (E + 15) / 16;
    const int nbEdge   = (numTiles + 7) / 8;
    const int nbAtt    = (C + 255) / 256;

    for (int l = 0; l < L; ++l) {
        float* vf_next = (l == L - 1) ? outV : ((l == 0) ? vfA : vfB);
        float* cf_next = (l == L - 1) ? outC : ((l == 0) ? cfA : cfB);

        // attention + softmax denominator (deterministic tree reduce)
        att_kernel<<<nbAtt, 256, 0, stream>>>(
            cf_cur, att_w1 + l * 192, att_b1 + l * 64, att_w2 + l * 64,
            att_b2 + l, attExp, partials, C);
        reduce_inv_kernel<<<1, 256, 0, stream>>>(partials, nbAtt, invS);

        // var -> clause
        zero_kernel<<<2048, 256, 0, stream>>>((float4*)msg, (size_t)C * 16);
        edge_mlp_scatter_kernel<<<nbEdge, 256, 0, stream>>>(
            vf_cur, var_idx, ef,
            v2c_w1 + l * 320, v2c_b1 + l * 64, v2c_w2 + l * 4096, v2c_b2 + l * 64,
            attExp, invS, cls_idx, msg, E, numTiles);
        gru_kernel<<<(C + 255) / 256, 256, 0, stream>>>(
            msg, cf_cur, cls_wih + l * 576, cls_whh + l * 27,
            cls_bih + l * 9, cls_bhh + l * 9, cf_next, C);

        // clause -> var (uses OLD cf)
        zero_kernel<<<2048, 256, 0, stream>>>((float4*)msg, (size_t)V * 16);
        edge_mlp_scatter_kernel<<<nbEdge, 256, 0, stream>>>(
            cf_cur, cls_idx, ef,
            c2v_w1 + l * 320, c2v_b1 + l * 64, c2v_w2 + l * 4096, c2v_b2 + l * 64,
            nullptr, invS, var_idx, msg, E, numTiles);
        gru_kernel<<<(V + 255) / 256, 256, 0, stream>>>(
            msg, vf_cur, var_wih + l * 576, var_whh + l * 27,
            var_bih + l * 9, var_bhh + l * 9, vf_next, V);

        vf_cur = vf_next;
        cf_cur = cf_next;
    }
}